// LinearEncoder_6760278524376
// MI455X (gfx1250) — compile-verified
//
#include <hip/hip_runtime.h>
#include <hip/hip_bf16.h>

typedef __attribute__((ext_vector_type(2))) float v2f;
typedef __attribute__((ext_vector_type(4))) float v4f;
typedef __attribute__((ext_vector_type(8))) float v8f;

#define D 128  // D_IN == D_OUT == 128

// ---------------------------------------------------------------- degrees ---
__global__ __launch_bounds__(256) void init_deg_kernel(float* __restrict__ deg, int n) {
    int i = blockIdx.x * blockDim.x + threadIdx.x;
    if (i < n) deg[i] = 1.0f;  // self-loop contribution folded in
}

__global__ __launch_bounds__(256) void count_deg_kernel(const int* __restrict__ ei, int E,
                                                        float* __restrict__ deg) {
    int e = blockIdx.x * blockDim.x + threadIdx.x;
    if (e < E) {
        int dst = ei[E + e];  // edge_index[1][e]
        atomicAdd(&deg[dst], 1.0f);
    }
}

__global__ __launch_bounds__(256) void rsqrt_deg_kernel(float* __restrict__ deg, int n) {
    int i = blockIdx.x * blockDim.x + threadIdx.x;
    if (i < n) {
        float d = deg[i];
        deg[i] = (d > 0.0f) ? rsqrtf(d) : 0.0f;
    }
}

// ------------------------------------------------------- h = x @ W^T (WMMA) --
// Exact fp32 via V_WMMA_F32_16X16X4_F32. Block = 8 waves; block owns one
// 16-row M tile, wave w owns 16-col N tile w (8*16 = 128 = D_OUT).
__global__ __launch_bounds__(256) void gemm_xwT_kernel(const float* __restrict__ x,
                                                       const float* __restrict__ w,
                                                       float* __restrict__ h, int n) {
    const int lane = threadIdx.x & 31;
    const int wv   = threadIdx.x >> 5;         // 0..7 -> N tile
    const int m0   = blockIdx.x * 16;
    const int n0   = wv * 16;
    const int l15  = lane & 15;
    const int kh   = (lane >> 4) << 1;         // lanes 0-15: K+0/K+1; lanes 16-31: K+2/K+3

    int row = m0 + l15;
    if (row >= n) row = n - 1;                 // clamp loads (keeps EXEC all-1 for WMMA)
    const int col = n0 + l15;                  // B[k][n] = W[n][k] (h = x * W^T)

    const float* __restrict__ xr = x + (size_t)row * D;
    const float* __restrict__ wr = w + (size_t)col * D;

    v8f acc = {};
    #pragma unroll
    for (int k = 0; k < D; k += 4) {
        v2f a = *(const v2f*)(xr + k + kh);    // A 16x4: VGPR0=K+kh, VGPR1=K+kh+1
        v2f b = *(const v2f*)(wr + k + kh);    // B 4x16: same K striping, N=lane&15
        acc = __builtin_amdgcn_wmma_f32_16x16x4_f32(
            /*neg_a=*/false, a, /*neg_b=*/false, b,
            /*c_mod=*/(short)0, acc, /*reuse_a=*/false, /*reuse_b=*/false);
    }

    // C/D layout: VGPR r -> M = r (lanes 0-15) or r+8 (lanes 16-31), N = lane&15
    const int mb = m0 + ((lane >> 4) << 3);
    #pragma unroll
    for (int r = 0; r < 8; ++r) {
        int m = mb + r;
        if (m < n) h[(size_t)m * D + n0 + l15] = acc[r];
    }
}

// --------------------------------------- out = bias + dis^2 * h (self loop) --
__global__ __launch_bounds__(256) void init_out_kernel(const float* __restrict__ h,
                                                       const float* __restrict__ dis,
                                                       const float* __restrict__ bias,
                                                       float* __restrict__ out,
                                                       int n, int total) {
    int idx = blockIdx.x * blockDim.x + threadIdx.x;
    if (idx >= total) return;
    int nd = n * D;
    if (idx < nd) {
        int node = idx >> 7;    // / 128
        int f    = idx & 127;
        float s  = dis[node];
        out[idx] = bias[f] + s * s * h[idx];
    } else {
        out[idx] = 0.0f;        // second tuple output (scalar 0)
    }
}

// ------------------------------- edge aggregation: out[dst] += norm*h[src] ---
// One wave32 per edge; each lane moves 4 consecutive floats (float4 load +
// 4x global_atomic_add_f32). h/out/dis are all L2-resident (103 MB < 192 MB).
__global__ __launch_bounds__(256) void edge_agg_kernel(const int* __restrict__ ei, int E,
                                                       const float* __restrict__ h,
                                                       const float* __restrict__ dis,
                                                       float* __restrict__ out) {
    const int lane  = threadIdx.x & 31;
    int wave   = (blockIdx.x * blockDim.x + threadIdx.x) >> 5;
    int nwaves = (gridDim.x * blockDim.x) >> 5;
    for (int e = wave; e < E; e += nwaves) {
        int src = ei[e];
        int dst = ei[E + e];
        float wnorm = dis[src] * dis[dst];
        v4f hv = *(const v4f*)(h + (size_t)src * D + lane * 4);
        float* op = out + (size_t)dst * D + lane * 4;
        atomicAdd(op + 0, wnorm * hv.x);
        atomicAdd(op + 1, wnorm * hv.y);
        atomicAdd(op + 2, wnorm * hv.z);
        atomicAdd(op + 3, wnorm * hv.w);
    }
}

// ---------------------------------------------------------------- launcher ---
extern "C" void kernel_launch(void* const* d_in, const int* in_sizes, int n_in,
                              void* d_out, int out_size, void* d_ws, size_t ws_size,
                              hipStream_t stream) {
    const float* x    = (const float*)d_in[0];
    const int*   ei   = (const int*)d_in[1];
    const float* w    = (const float*)d_in[2];
    const float* bias = (const float*)d_in[3];
    float*       out  = (float*)d_out;

    const int n = in_sizes[0] / D;   // 100000 nodes
    const int E = in_sizes[1] / 2;   // 3200000 edges

    // workspace: h [n*D floats] | dis [n floats]
    float* h   = (float*)d_ws;
    float* dis = h + (size_t)n * D;

    int tpb = 256;
    init_deg_kernel<<<(n + tpb - 1) / tpb, tpb, 0, stream>>>(dis, n);
    count_deg_kernel<<<(E + tpb - 1) / tpb, tpb, 0, stream>>>(ei, E, dis);
    rsqrt_deg_kernel<<<(n + tpb - 1) / tpb, tpb, 0, stream>>>(dis, n);
    gemm_xwT_kernel<<<(n + 15) / 16, tpb, 0, stream>>>(x, w, h, n);
    init_out_kernel<<<(out_size + tpb - 1) / tpb, tpb, 0, stream>>>(h, dis, bias, out, n, out_size);
    edge_agg_kernel<<<4096, tpb, 0, stream>>>(ei, E, h, dis, out);
}